// MMPhysicsField_56349970923944
// MI455X (gfx1250) — compile-verified
//
#include <hip/hip_runtime.h>
#include <hip/hip_bf16.h>

#define TILE 512
#define KE   14.3996f
#define EPSF 1e-8f

// ---------------------------------------------------------------------------
// Partial kernel: block (x, cy, b) handles QM atoms [x*256, x*256+256) of
// batch b against MM chunk cy (mchunk MM atoms). MM tile is staged into LDS
// with CDNA5 async global->LDS b128 copies, then repacked to float4
// {x, y, z, KE*q*mask} so the inner loop is one broadcast ds_load_b128 per j.
// Writes float4{V,Ex,Ey,Ez} partials to ws (or directly to out if nch==1).
// ---------------------------------------------------------------------------
__global__ __launch_bounds__(256) void mm_field_partial(
    const float* __restrict__ qm,     // [N,3]
    const float* __restrict__ mmp,    // [B,M,3]
    const float* __restrict__ mmq,    // [B,M]
    const int*   __restrict__ nnode,  // [B]
    const int*   __restrict__ nmm,    // [B]
    float*       __restrict__ ws,     // [nch,N,4] partials
    float*       __restrict__ out,    // direct-mode output [N + 3N]
    int N, int M, int mchunk, int direct)
{
    __shared__ __align__(16) float  stage[TILE * 3];  // raw positions (AoS)
    __shared__ __align__(16) float4 tile[TILE];       // packed {x,y,z,KE*q}

    const int b  = blockIdx.z;
    const int cy = blockIdx.y;

    // prefix sum of num_nodes to find this batch's atom range (B is tiny)
    int start = 0;
    for (int i = 0; i < b; ++i) start += nnode[i];
    const int nb = nnode[b];

    if ((int)(blockIdx.x * blockDim.x) >= nb) return;  // uniform exit

    const int  t      = blockIdx.x * blockDim.x + threadIdx.x;
    const bool active = (t < nb);
    const int  atom   = start + t;
    const int  Mb     = nmm[b];

    float qx = 0.f, qy = 0.f, qz = 0.f;
    if (active) {
        qx = qm[3 * (size_t)atom + 0];
        qy = qm[3 * (size_t)atom + 1];
        qz = qm[3 * (size_t)atom + 2];
    }

    const float* gp = mmp + (size_t)b * M * 3;
    const float* gq = mmq + (size_t)b * M;

    const int j0 = cy * mchunk;
    const int j1 = (j0 + mchunk < M) ? (j0 + mchunk) : M;

    float accV = 0.f, accX = 0.f, accY = 0.f, accZ = 0.f;

    // LDS byte offset of stage (flat addr low 32 bits == LDS offset)
    const unsigned stage_base = (unsigned)(size_t)(&stage[0]);

    for (int tb = j0; tb < j1; tb += TILE) {
        // ---- stage positions: async global->LDS b128 when tile is full ----
        if (tb + TILE <= M) {
            const char* gsrc = (const char*)(gp + (size_t)tb * 3);
            for (int i = threadIdx.x; i < (TILE * 3 * 4) / 16; i += 256) {
                unsigned    l  = stage_base + (unsigned)(i * 16);
                const char* ga = gsrc + (size_t)i * 16;
                asm volatile("global_load_async_to_lds_b128 %0, %1, off"
                             :: "v"(l), "v"(ga) : "memory");
            }
        } else {
            for (int i = threadIdx.x; i < TILE * 3; i += 256) {
                int j = tb + i / 3;
                stage[i] = (j < M) ? gp[(size_t)tb * 3 + i] : 0.0f;
            }
        }

        asm volatile("s_wait_asynccnt 0x0" ::: "memory");
        __syncthreads();

        // ---- repack: {x,y,z, KE*q*mask} per MM atom (mask folded here) ----
        for (int i = threadIdx.x; i < TILE; i += 256) {
            int   j = tb + i;
            float w = (j < Mb && j < j1) ? (KE * gq[j]) : 0.0f;
            tile[i] = make_float4(stage[3 * i + 0],
                                  stage[3 * i + 1],
                                  stage[3 * i + 2], w);
        }
        __syncthreads();

        if (active) {
            #pragma unroll 8
            for (int j = 0; j < TILE; ++j) {
                float4 p = tile[j];              // one broadcast ds_load_b128
                float dx = qx - p.x;
                float dy = qy - p.y;
                float dz = qz - p.z;
                float d2  = __builtin_fmaf(dx, dx,
                            __builtin_fmaf(dy, dy, dz * dz));
                float d   = __builtin_amdgcn_sqrtf(d2);
                float inv = __builtin_amdgcn_rcpf(d + EPSF);
                float wi  = p.w * inv;           // w / d
                accV += wi;
                float wi3 = wi * inv * inv;      // w / d^3
                accX = __builtin_fmaf(wi3, dx, accX);
                accY = __builtin_fmaf(wi3, dy, accY);
                accZ = __builtin_fmaf(wi3, dz, accZ);
            }
        }
        __syncthreads();
    }

    if (active) {
        if (direct) {
            out[atom]             = accV;
            out[N + 3 * atom + 0] = accX;
            out[N + 3 * atom + 1] = accY;
            out[N + 3 * atom + 2] = accZ;
        } else {
            float4* w4 = (float4*)ws;
            w4[(size_t)cy * N + atom] = make_float4(accV, accX, accY, accZ);
        }
    }
}

// ---------------------------------------------------------------------------
// Deterministic chunk reduction: out = [V(N) | E(N,3)]
// ---------------------------------------------------------------------------
__global__ __launch_bounds__(256) void mm_field_reduce(
    const float4* __restrict__ ws, float* __restrict__ out, int N, int nch)
{
    int i = blockIdx.x * blockDim.x + threadIdx.x;
    if (i >= N) return;
    float V = 0.f, x = 0.f, y = 0.f, z = 0.f;
    for (int c = 0; c < nch; ++c) {
        float4 p = ws[(size_t)c * N + i];
        V += p.x; x += p.y; y += p.z; z += p.w;
    }
    out[i]             = V;
    out[N + 3 * i + 0] = x;
    out[N + 3 * i + 1] = y;
    out[N + 3 * i + 2] = z;
}

extern "C" void kernel_launch(void* const* d_in, const int* in_sizes, int n_in,
                              void* d_out, int out_size, void* d_ws, size_t ws_size,
                              hipStream_t stream) {
    const float* qm    = (const float*)d_in[0];   // [N,3]
    const float* mmp   = (const float*)d_in[1];   // [B,M,3]
    const float* mmq   = (const float*)d_in[2];   // [B,M]
    const int*   nnode = (const int*)d_in[3];     // [B]
    const int*   nmm   = (const int*)d_in[4];     // [B]
    float*       out   = (float*)d_out;

    const int N = in_sizes[0] / 3;
    const int B = in_sizes[3];
    const int M = in_sizes[2] / B;

    // Split M into chunks for parallelism; shrink chunk count if ws too small.
    const size_t per_chunk_bytes = (size_t)N * 4 * sizeof(float);
    int mchunk = TILE;
    int nch    = (M + mchunk - 1) / mchunk;
    while (nch > 1 && (size_t)nch * per_chunk_bytes > ws_size) {
        mchunk *= 2;
        nch = (M + mchunk - 1) / mchunk;
    }
    const int direct = (nch == 1);

    dim3 grid((N + 255) / 256, nch, B);
    mm_field_partial<<<grid, 256, 0, stream>>>(
        qm, mmp, mmq, nnode, nmm, (float*)d_ws, out, N, M, mchunk, direct);

    if (!direct) {
        mm_field_reduce<<<(N + 255) / 256, 256, 0, stream>>>(
            (const float4*)d_ws, out, N, nch);
    }
}